// GCNCheb_19662360281153
// MI455X (gfx1250) — compile-verified
//
#include <hip/hip_runtime.h>

typedef __attribute__((ext_vector_type(2))) float v2f;
typedef __attribute__((ext_vector_type(8))) float v8f;

#define C_DIM 128

// ---------------------------------------------------------------- utility ---
__global__ void fill_zero_kernel(float* __restrict__ p, int n) {
    int i = blockIdx.x * blockDim.x + threadIdx.x;
    if (i < n) p[i] = 0.0f;
}

// out[i] = -in[i]  (in == out allowed)
__global__ void negate_kernel(const float* __restrict__ in, float* __restrict__ out, int n) {
    int i = blockIdx.x * blockDim.x + threadIdx.x;
    if (i < n) out[i] = -in[i];
}

// ------------------------------------------------------------------ SpMM ----
// y[row] += scale * lap[e] * x[col]; one wave32 per edge, lane covers 4 chans.
__global__ __launch_bounds__(256)
void spmm_atomic_kernel(const long long* __restrict__ ei,   // [2, E]
                        const float* __restrict__ lap,      // [E]
                        const float* __restrict__ xin,      // [N, 128]
                        float* __restrict__ yout,           // [N, 128]
                        float scale, int nEdges) {
    int gtid = blockIdx.x * blockDim.x + threadIdx.x;
    int e    = gtid >> 5;           // wave id == edge id
    int lane = threadIdx.x & 31;
    if (e >= nEdges) return;

    long long r = ei[e];
    long long c = ei[(long long)nEdges + e];
    float w = lap[e] * scale;

    const float4* xv = (const float4*)(xin + c * C_DIM);
    float4 v = xv[lane];            // 32 lanes x 4 floats = 128 channels

    float* yo = yout + r * C_DIM + lane * 4;
    atomicAdd(yo + 0, w * v.x);
    atomicAdd(yo + 1, w * v.y);
    atomicAdd(yo + 2, w * v.z);
    atomicAdd(yo + 3, w * v.w);
}

// ------------------------------------------------------------------ GEMM ----
// out[tile] = (accum ? accum : 0) + X0*W0 + (X1 ? X1*W1 : 0) + (bias ? bias : 0)
// Block: 256 threads = 8 waves; block handles a 64-row slab, wave w handles
// column tile [16w, 16w+16). Each wave: 4 sub-tiles of 16 rows, f32 WMMA.
__global__ __launch_bounds__(256)
void cheb_gemm_kernel(const float* __restrict__ X0, const float* __restrict__ W0,
                      const float* __restrict__ X1, const float* __restrict__ W1,
                      const float* __restrict__ bias,
                      const float* __restrict__ accum,
                      float* __restrict__ out, int nRows) {
    const int waveId = threadIdx.x >> 5;    // 0..7 -> column tile
    const int lane   = threadIdx.x & 31;
    const int half   = lane >> 4;           // 0: K=k0,k0+1 | 1: K=k0+2,k0+3
    const int l16    = lane & 15;
    const int row0   = blockIdx.x * 64;
    const int col0   = waveId * 16;

    v8f acc[4];
    acc[0] = v8f{}; acc[1] = v8f{}; acc[2] = v8f{}; acc[3] = v8f{};

    const float* Xs[2] = {X0, X1};
    const float* Ws[2] = {W0, W1};

    for (int m = 0; m < 2; ++m) {
        const float* X = Xs[m];
        const float* W = Ws[m];
        if (X == nullptr) continue;
        for (int k0 = 0; k0 < C_DIM; k0 += 4) {
            const int kk = k0 + 2 * half;
            // B fragment: B[kk][col0+l16], B[kk+1][col0+l16]
            v2f b;
            b.x = W[kk * C_DIM + col0 + l16];
            b.y = W[(kk + 1) * C_DIM + col0 + l16];
#pragma unroll
            for (int st = 0; st < 4; ++st) {
                int r = row0 + st * 16 + l16;
                if (r >= nRows) r = nRows - 1;   // clamp (stores are guarded)
                v2f a = *(const v2f*)(X + (long long)r * C_DIM + kk);
                acc[st] = __builtin_amdgcn_wmma_f32_16x16x4_f32(
                    /*neg_a=*/false, a, /*neg_b=*/false, b,
                    /*c_mod=*/(short)0, acc[st],
                    /*reuse_a=*/false, /*reuse_b=*/false);
            }
        }
    }

    const float bv = bias ? bias[col0 + l16] : 0.0f;
#pragma unroll
    for (int st = 0; st < 4; ++st) {
#pragma unroll
        for (int i = 0; i < 8; ++i) {
            int r = row0 + st * 16 + half * 8 + i;   // C/D layout: M = i + 8*half
            if (r < nRows) {
                long long idx = (long long)r * C_DIM + col0 + l16;
                float v = acc[st][i] + bv;
                if (accum) v += accum[idx];
                out[idx] = v;
            }
        }
    }
}

// ---------------------------------------------------------------- launch ----
extern "C" void kernel_launch(void* const* d_in, const int* in_sizes, int n_in,
                              void* d_out, int out_size, void* d_ws, size_t ws_size,
                              hipStream_t stream) {
    const float*     x      = (const float*)d_in[0];       // [N,128]
    const float*     lap    = (const float*)d_in[1];       // [E]
    const float*     weight = (const float*)d_in[2];       // [4,128,128]
    const float*     bias   = (const float*)d_in[3];       // [128]
    const long long* ei     = (const long long*)d_in[4];   // [2,E] int64
    float*           out    = (float*)d_out;

    const int nNodes = in_sizes[0] / C_DIM;
    const int nEdges = in_sizes[1];
    const size_t nElem = (size_t)nNodes * C_DIM;

    float* bufA = (float*)d_ws;                 // Tx1, later Tx3
    float* bufB = bufA + nElem;                 // Tx2

    const float* W0 = weight + 0 * C_DIM * C_DIM;
    const float* W1 = weight + 1 * C_DIM * C_DIM;
    const float* W2 = weight + 2 * C_DIM * C_DIM;
    const float* W3 = weight + 3 * C_DIM * C_DIM;

    const int elemBlocks = (int)((nElem + 255) / 256);
    const int edgeBlocks = (nEdges + 7) / 8;        // 8 waves (=edges) per block
    const int rowBlocks  = (nNodes + 63) / 64;

    // Tx1 = L x
    fill_zero_kernel<<<elemBlocks, 256, 0, stream>>>(bufA, (int)nElem);
    spmm_atomic_kernel<<<edgeBlocks, 256, 0, stream>>>(ei, lap, x, bufA, 1.0f, nEdges);

    // out = x W0 + Tx1 W1 + bias
    cheb_gemm_kernel<<<rowBlocks, 256, 0, stream>>>(x, W0, bufA, W1, bias, nullptr, out, nNodes);

    // Tx2 = 2 L Tx1 - x
    negate_kernel<<<elemBlocks, 256, 0, stream>>>(x, bufB, (int)nElem);
    spmm_atomic_kernel<<<edgeBlocks, 256, 0, stream>>>(ei, lap, bufA, bufB, 2.0f, nEdges);

    // out += Tx2 W2
    cheb_gemm_kernel<<<rowBlocks, 256, 0, stream>>>(bufB, W2, nullptr, nullptr, nullptr, out, out, nNodes);

    // Tx3 = 2 L Tx2 - Tx1   (in place over bufA)
    negate_kernel<<<elemBlocks, 256, 0, stream>>>(bufA, bufA, (int)nElem);
    spmm_atomic_kernel<<<edgeBlocks, 256, 0, stream>>>(ei, lap, bufB, bufA, 2.0f, nEdges);

    // out += Tx3 W3
    cheb_gemm_kernel<<<rowBlocks, 256, 0, stream>>>(bufA, W3, nullptr, nullptr, nullptr, out, out, nNodes);
}